// MF_27436251087258
// MI455X (gfx1250) — compile-verified
//
#include <hip/hip_runtime.h>

typedef float v2f __attribute__((ext_vector_type(2)));
typedef float v8f __attribute__((ext_vector_type(8)));

// One wave (32 lanes) handles 16 batch rows.
// Phase 1: corr[16x64] = A_gathered[16x16] @ B[16x64] via 16x v_wmma_f32_16x16x4_f32.
// Phase 2: fuse item_emb/user_emb gathers + dot product on the accumulator
//          registers, reduce across the 16 lanes of each half-wave, add biases.
__global__ __launch_bounds__(32) void mf_fused_wmma_kernel(
    const int*   __restrict__ users,
    const int*   __restrict__ items,
    const float* __restrict__ user_emb,   // [NUM_USERS, 64]
    const float* __restrict__ item_emb,   // [NUM_ITEMS, 64]
    const float* __restrict__ user_bias,  // [NUM_USERS]
    const float* __restrict__ item_bias,  // [NUM_ITEMS]
    const float* __restrict__ A,          // [NUM_ITEMS, 16]
    const float* __restrict__ Bm,         // [16, 64]
    float*       __restrict__ out)        // [BATCH]
{
    const int lane = threadIdx.x & 31;
    const int half = lane >> 4;      // 0: lanes 0-15, 1: lanes 16-31
    const int n    = lane & 15;
    const int base = blockIdx.x << 4; // 16 rows per wave

    // Lane n holds the indices for batch row (base + n); identical in both halves.
    const int itemIdx = items[base + n];
    const int userIdx = users[base + n];

    // A-fragment for V_WMMA_F32_16X16X4_F32 (16x4, MxK):
    //   lanes 0-15:  M = n, VGPR0/1 = K = k0, k0+1
    //   lanes 16-31: M = n, VGPR0/1 = K = k0+2, k0+3
    v2f afrag[4];
    {
        const float* arow = A + (long)itemIdx * 16 + half * 2;
        #pragma unroll
        for (int kc = 0; kc < 4; ++kc) {
            afrag[kc].x = arow[kc * 4 + 0];
            afrag[kc].y = arow[kc * 4 + 1];
        }
    }

    // corr = A_g @ B : four 16x16 f32 output tiles (cols t*16 .. t*16+15),
    // each accumulated over K=16 with 4 chained K=4 WMMAs.
    // B-fragment (4x16, KxN) mirrors A: lane%16 = N, VGPR/half select K.
    v8f acc[4];
    #pragma unroll
    for (int t = 0; t < 4; ++t) {
        v8f c = {};
        #pragma unroll
        for (int kc = 0; kc < 4; ++kc) {
            const int k0 = kc * 4 + half * 2;
            v2f bfrag;
            bfrag.x = Bm[(k0 + 0) * 64 + t * 16 + n];
            bfrag.y = Bm[(k0 + 1) * 64 + t * 16 + n];
            c = __builtin_amdgcn_wmma_f32_16x16x4_f32(
                    /*neg_a=*/false, afrag[kc],
                    /*neg_b=*/false, bfrag,
                    /*c_mod=*/(short)0, c,
                    /*reuse_a=*/false, /*reuse_b=*/false);
        }
        acc[t] = c;
    }

    // C/D layout: VGPR v, lane (half,n) holds corr[M = v + 8*half][N = t*16 + n].
    // Fuse the two embedding gathers and the dot product per row.
    int   itArr[8], usArr[8];
    float s[8];
    #pragma unroll
    for (int v = 0; v < 8; ++v) {
        const int r  = v + 8 * half;          // batch row within tile
        itArr[v] = __shfl(itemIdx, r, 32);    // items[base + r]
        usArr[v] = __shfl(userIdx, r, 32);    // users[base + r]
        const float* ie = item_emb + (long)itArr[v] * 64 + n;
        const float* ue = user_emb + (long)usArr[v] * 64 + n;
        float p = 0.0f;
        #pragma unroll
        for (int t = 0; t < 4; ++t) {
            p += (ie[t * 16] + acc[t][v]) * ue[t * 16];
        }
        s[v] = p;
    }

    // Reduce across the 16 lanes of each half (xor masks keep bit4 fixed).
    #pragma unroll
    for (int v = 0; v < 8; ++v) {
        float x = s[v];
        x += __shfl_xor(x, 8, 32);
        x += __shfl_xor(x, 4, 32);
        x += __shfl_xor(x, 2, 32);
        x += __shfl_xor(x, 1, 32);
        s[v] = x;
    }

    // Lane 0 of each half writes its 8 predictions with biases.
    if (n == 0) {
        #pragma unroll
        for (int v = 0; v < 8; ++v) {
            out[base + v + 8 * half] =
                s[v] + user_bias[usArr[v]] + item_bias[itArr[v]];
        }
    }
}

extern "C" void kernel_launch(void* const* d_in, const int* in_sizes, int n_in,
                              void* d_out, int out_size, void* d_ws, size_t ws_size,
                              hipStream_t stream) {
    const int*   users     = (const int*)  d_in[0];
    const int*   items     = (const int*)  d_in[1];
    const float* user_emb  = (const float*)d_in[2];
    const float* item_emb  = (const float*)d_in[3];
    const float* user_bias = (const float*)d_in[4];
    const float* item_bias = (const float*)d_in[5];
    const float* A         = (const float*)d_in[6];
    const float* Bm        = (const float*)d_in[7];
    float* out = (float*)d_out;

    const int batch  = in_sizes[0];     // 16384
    const int blocks = batch >> 4;      // 16 rows per wave, 1 wave per block

    mf_fused_wmma_kernel<<<blocks, 32, 0, stream>>>(
        users, items, user_emb, item_emb, user_bias, item_bias, A, Bm, out);
}